// RPNPooling_7352984011596
// MI455X (gfx1250) — compile-verified
//
#include <hip/hip_runtime.h>

// RPN/ROI bilinear pooling for MI455X (gfx1250).
// B=2, N=512 ROIs/batch, feature map (2,128,128,256) f32, pool=14.
// Strategy: one workgroup per ROI. ROI source footprint is <= 15x15 pixels
// (roi extents are in [8,15]); stage it into LDS via CDNA5 async global->LDS
// DMA (global_load_async_to_lds_b128 / ASYNCcnt), then do the bilinear blend
// out of LDS. Channels processed in 4 passes of 64 so the LDS patch is
// 57.6 KB. Output stores are non-temporal (write-once 205 MB stream) so the
// 192 MB L2 stays reserved for the feature map (read ~7x per byte).

#define B_    2
#define N_    512
#define H_    128
#define W_    128
#define C_    256
#define PP    14           // pool
#define NPASS 4
#define K4P   16           // float4-channels per pass (16*4 = 64 channels)
#define MAXP  15           // max patch dimension
#define PATCH_BYTES (MAXP * MAXP * K4P * 16)   // 57600

typedef float v4f __attribute__((ext_vector_type(4)));   // 16B aligned

__global__ __launch_bounds__(256) void RPNPooling_kernel(
    const float* __restrict__ feat,
    const int*   __restrict__ roi,
    float*       __restrict__ out)
{
    // No static __shared__ anywhere: dynamic LDS region starts at offset 0,
    // so raw byte offsets given to the async DMA match the generic-pointer
    // reads below. Tables live at the tail of the dynamic region.
    extern __shared__ __align__(16) char dynsm[];
    v4f*   patch = (v4f*)dynsm;
    int*   s_r0  = (int*)(dynsm + PATCH_BYTES);
    int*   s_r1  = s_r0 + PP;
    int*   s_c0  = s_r1 + PP;
    int*   s_c1  = s_c0 + PP;
    float* s_fr  = (float*)(s_c1 + PP);
    float* s_fc  = s_fr + PP;

    const int ri  = blockIdx.x;          // 0..1023
    const int tid = threadIdx.x;         // 0..255
    const int b   = ri >> 9;             // ri / N_

    // roi row: [y1, x1, y2, x2]; reference maps x->rows(H axis), y->cols(W axis)
    const int y1 = roi[ri * 4 + 0];
    const int x1 = roi[ri * 4 + 1];
    const int y2 = roi[ri * 4 + 2];
    const int x2 = roi[ri * 4 + 3];
    const int sizeR = x2 - x1;           // rows, in [8,15]
    const int sizeC = y2 - y1;           // cols, in [8,15]

    // Per-axis legacy-resize tables (exactly mirrors reference arithmetic:
    // local = (i * size) / 14.0 with true fp32 division).
    if (tid < 2 * PP) {
        const bool rAxis = tid < PP;
        const int  i  = rAxis ? tid : tid - PP;
        const int  sz = rAxis ? sizeR : sizeC;
        const float local = (float)(i * sz) / 14.0f;
        int   i0 = (int)floorf(local);
        int   i1 = (i0 + 1 < sz - 1) ? (i0 + 1) : (sz - 1);   // min
        float fr = local - (float)i0;
        if (rAxis) { s_r0[i] = i0; s_r1[i] = i1; s_fr[i] = fr; }
        else       { s_c0[i] = i0; s_c1[i] = i1; s_fc[i] = fr; }
    }

    const unsigned long long featBase = (unsigned long long)feat;
    const int kk     = tid & (K4P - 1);  // float4 channel within pass
    const int pStart = tid >> 4;         // 0..15, pixel cursor
    const int stageN = sizeC * K4P;      // lanes that stage (128..240)

    for (int pass = 0; pass < NPASS; ++pass) {
        if (pass) __syncthreads();       // previous compute done before overwrite

        // ---- stage patch rows via CDNA5 async global->LDS DMA -------------
        if (tid < stageN) {
            const int t  = tid >> 4;     // patch column
            const int kc = tid & (K4P - 1);
            unsigned ldsOff = (unsigned)((t * K4P + kc) * 16);
            unsigned gOff   = (unsigned)((((b * H_ + x1) * W_ + (y1 + t)) * C_
                                          + (pass * K4P + kc) * 4) * 4);
            const unsigned ldsRowStride = (unsigned)(sizeC * K4P * 16);
            const unsigned gRowStride   = (unsigned)(W_ * C_ * 4);
            for (int s = 0; s < sizeR; ++s) {
                asm volatile(
                    "global_load_async_to_lds_b128 %0, %1, %2"
                    :: "v"(ldsOff), "v"(gOff), "s"(featBase)
                    : "memory");
                ldsOff += ldsRowStride;
                gOff   += gRowStride;
            }
        }
        asm volatile("s_wait_asynccnt 0" ::: "memory");
        __syncthreads();

        // ---- bilinear blend out of LDS ------------------------------------
        const int chBase = pass * K4P;
        for (int p = pStart; p < PP * PP; p += 16) {
            const int pr = p / PP;
            const int pc = p - pr * PP;
            const int r0 = s_r0[pr], r1 = s_r1[pr];
            const int c0 = s_c0[pc], c1 = s_c1[pc];
            const float fr = s_fr[pr], fc = s_fc[pc];
            const float gr = 1.0f - fr, gc = 1.0f - fc;

            const v4f v00 = patch[(r0 * sizeC + c0) * K4P + kk];  // ds_load_b128
            const v4f v01 = patch[(r0 * sizeC + c1) * K4P + kk];
            const v4f v10 = patch[(r1 * sizeC + c0) * K4P + kk];
            const v4f v11 = patch[(r1 * sizeC + c1) * K4P + kk];

            // Same op order as reference: top/bot column lerp, then row lerp.
            const v4f o = (v00 * gc + v01 * fc) * gr + (v10 * gc + v11 * fc) * fr;

            v4f* dst = (v4f*)(out + (((size_t)ri * (PP * PP) + p) * C_
                                     + (size_t)(chBase + kk) * 4));
            __builtin_nontemporal_store(o, dst);   // NT: keep L2 for features
        }
    }
}

extern "C" void kernel_launch(void* const* d_in, const int* in_sizes, int n_in,
                              void* d_out, int out_size, void* d_ws, size_t ws_size,
                              hipStream_t stream) {
    const float* feat = (const float*)d_in[0];   // (2,128,128,256) f32
    const int*   roi  = (const int*)d_in[1];     // (2,512,4) i32
    float*       out  = (float*)d_out;           // (1024,14,14,256) f32

    const size_t shmem = (size_t)PATCH_BYTES + 6 * PP * sizeof(int); // 57936 B
    dim3 grid(B_ * N_);
    dim3 block(256);
    RPNPooling_kernel<<<grid, block, shmem, stream>>>(feat, roi, out);
}